// MultiLabelHeadOnlyTask_63806034149584
// MI455X (gfx1250) — compile-verified
//
#include <hip/hip_runtime.h>

typedef __attribute__((ext_vector_type(2))) float v2f;
typedef __attribute__((ext_vector_type(8))) float v8f;

#define NPS          6                 // non-task flag columns
#define BATCH        65536             // B
#define FDIM         128               // F (GEMM K-reduction)
#define KHEADS       64                // K heads
#define CDIM         8                 // C
#define NDIM         (KHEADS * CDIM)   // 512 (GEMM N)
#define FLAGS_STRIDE (NPS + KHEADS)    // 70
#define WAVES_PER_BLOCK 8
#define ROWS_PER_BLOCK  (WAVES_PER_BLOCK * 16)   // 128 rows per block
#define NT           4                 // 16-wide n-tiles per wave (64 columns)
#define HEADS_PER_BLOCK (NT * 16 / CDIM)         // 8 heads per block
#define COLS_PER_BLOCK  (NT * 16)                // 64 columns per block
#define LDS_STRIDE   (FDIM + 4)       // 132 floats: bank-conflict-free pad

// Each wave32 computes a 16x64 output strip: 4 n-tiles sharing one set of
// register-resident A fragments (32 x v2f = 64 VGPRs), 128 WMMAs per wave.
//   out = mask .* (x @ Bmat + bias),  Bmat[f, n] = W[n/8, f, n%8]
// fp32 WMMA keeps reference precision; problem is HBM/L2-bound so the fp32
// matrix path is the right tradeoff. N-blocking cuts x L2 traffic 4x.
__global__ __launch_bounds__(WAVES_PER_BLOCK * 32)
void head_gemm_kernel(const float* __restrict__ x,
                      const int*   __restrict__ flags,
                      const float* __restrict__ W,
                      const float* __restrict__ bias,
                      float*       __restrict__ out) {
    // Transposed B panel: Btile[col][f], stride 132 floats so hot-loop
    // ds_load_2addr_b64 reads hit all 64 banks uniquely.
    __shared__ float Btile[COLS_PER_BLOCK][LDS_STRIDE];   // ~34 KB

    const int n0 = blockIdx.y * COLS_PER_BLOCK;           // first column

    // Stage 8 heads of W (contiguous 8*128*8 floats) via float4 loads,
    // scattering the transpose into LDS.  8 x global_load_b128 per thread.
    {
        const float4* Wblk =
            (const float4*)(W + (size_t)blockIdx.y * (HEADS_PER_BLOCK * FDIM * CDIM));
        #pragma unroll
        for (int it = 0; it < 8; ++it) {
            const int i4 = threadIdx.x + it * (WAVES_PER_BLOCK * 32); // 0..2047
            const float4 v = Wblk[i4];
            const int i  = i4 * 4;              // element index in W block region
            const int kk = i >> 10;             // head within block (0..7)
            const int f  = (i >> 3) & (FDIM - 1);
            const int c  = i & 7;               // 0 or 4
            const int j  = kk * CDIM + c;       // column within block
            Btile[j + 0][f] = v.x;
            Btile[j + 1][f] = v.y;
            Btile[j + 2][f] = v.z;
            Btile[j + 3][f] = v.w;
        }
    }
    __syncthreads();

    const int wave = threadIdx.x >> 5;
    const int lane = threadIdx.x & 31;
    const int j    = lane & 15;              // row-in-tile / column-in-tile
    const int hi   = lane >> 4;              // which K-half this lane holds

    const int mbase = (blockIdx.x * WAVES_PER_BLOCK + wave) * 16;
    const float* xrow = x + (size_t)(mbase + j) * FDIM;
    __builtin_prefetch(xrow, 0, 0);          // global_prefetch_b8

    // A fragments: loaded once, reused for all 4 n-tiles.
    // Layout (16x4 fp32 A): lane(0-15) row=j K={4k+0,1}; lane(16-31) K={4k+2,3}
    v2f afrag[FDIM / 4];
    #pragma unroll
    for (int k0 = 0; k0 < FDIM / 4; ++k0) {
        afrag[k0] = *(const v2f*)(xrow + k0 * 4 + hi * 2);
    }

    #pragma unroll
    for (int t = 0; t < NT; ++t) {
        v8f acc = {0.f, 0.f, 0.f, 0.f, 0.f, 0.f, 0.f, 0.f};

        #pragma unroll
        for (int k0 = 0; k0 < FDIM / 4; ++k0) {
            const int koff = k0 * 4 + hi * 2;
            // B 4x16 layout: VGPR0=K even, VGPR1=K odd; hi lanes hold K+2
            v2f bb = *(const v2f*)(&Btile[t * 16 + j][koff]);
            acc = __builtin_amdgcn_wmma_f32_16x16x4_f32(
                /*neg_a=*/false, afrag[k0], /*neg_b=*/false, bb,
                /*c_mod=*/(short)0, acc, /*reuse_a=*/false, /*reuse_b=*/false);
        }

        // Epilogue: C/D layout VGPR r -> row = mbase + hi*8 + r, col = n.
        const int   n     = n0 + t * 16 + j;
        const float bn    = bias[n];         // b flat [K*C] == column index
        const int   khead = n >> 3;
        #pragma unroll
        for (int r = 0; r < 8; ++r) {
            const int   row = mbase + hi * 8 + r;
            const int   flg = flags[(size_t)row * FLAGS_STRIDE + NPS + khead];
            const float val = (flg == 1) ? (acc[r] + bn) : 0.0f;
            out[(size_t)row * NDIM + n] = val;
        }
    }
}

extern "C" void kernel_launch(void* const* d_in, const int* in_sizes, int n_in,
                              void* d_out, int out_size, void* d_ws, size_t ws_size,
                              hipStream_t stream) {
    const float* x     = (const float*)d_in[0];
    const int*   flags = (const int*)  d_in[1];
    const float* W     = (const float*)d_in[2];
    const float* b     = (const float*)d_in[3];
    float*       out   = (float*)d_out;

    dim3 grid(BATCH / ROWS_PER_BLOCK, NDIM / COLS_PER_BLOCK);   // (512, 8)
    dim3 block(WAVES_PER_BLOCK * 32);                           // 8 wave32
    hipLaunchKernelGGL(head_gemm_kernel, grid, block, 0, stream,
                       x, flags, W, b, out);
}